// SAGEClassifier_80470507258310
// MI455X (gfx1250) — compile-verified
//
#include <hip/hip_runtime.h>
#include <stdint.h>

// ---------------------------------------------------------------------------
// GraphSAGE 3-layer + BN/ReLU + linear head for MI455X (gfx1250, wave32).
//  - GEMMs: V_WMMA_F32_16X16X4_F32 (fp32 matrix pipe, matches fp32 reference)
//  - A panels staged global->LDS with the Tensor Data Mover (tensor_load_to_lds,
//    s_wait_tensorcnt), shared by all column-tile waves of the block.
//  - Aggregation: hardware global_atomic_add_f32 via unsafeAtomicAdd.
// ---------------------------------------------------------------------------

typedef __attribute__((ext_vector_type(2))) float v2f;
typedef __attribute__((ext_vector_type(8))) float v8f;
typedef __attribute__((ext_vector_type(4))) unsigned int v4u;
typedef __attribute__((ext_vector_type(8))) int v8i;
typedef __attribute__((ext_vector_type(4))) int v4i;

#define NN 50000
#define BN_EPS 1e-5f

// ---------------- TDM: 2D tile (rows x width f32, row stride = stride) ------
// Builds the D# per CDNA5 ISA:
//   group0: count=1 | lds_addr | global_addr[56:0] | type=2
//   group1: data_size=4B | tensor_dim0=width | tensor_dim1=rows |
//           tile_dim0=width | tile_dim1=rows | tensor_dim0_stride=stride
// This toolchain's builtin takes 6 args: (g0, g1, g2, g3, g_extra, cpol).
__device__ __forceinline__ void tdm_load_2d(void* lds_ptr, const void* gptr,
                                            int width, int rows, int stride) {
    unsigned long long ga = (unsigned long long)gptr;
    v4u g0;
    g0[0] = 1u;                                   // count=1, no gather
    g0[1] = (unsigned)(uintptr_t)lds_ptr;         // LDS byte address
    g0[2] = (unsigned)ga;                         // global_addr[31:0]
    g0[3] = ((unsigned)(ga >> 32) & 0x01FFFFFFu)  // global_addr[56:32]
            | (2u << 30);                         // type = 2 (image)
    v8i g1;
    g1[0] = 2 << 16;                              // data_size = 4 bytes
    g1[1] = (width & 0xFFFF) << 16;               // tensor_dim0[15:0]
    g1[2] = ((width >> 16) & 0xFFFF) | (rows << 16); // dim0[31:16] | dim1[15:0]
    g1[3] = (width & 0xFFFF) << 16;               // dim1[31:16]=0 | tile_dim0
    g1[4] = rows;                                 // tile_dim1 | tile_dim2=0
    g1[5] = stride;                               // tensor_dim0_stride[31:0]
    g1[6] = 0;                                    // stride hi | dim1_stride lo
    g1[7] = 0;
    v4i z4 = {0, 0, 0, 0};
    v8i z8 = {0, 0, 0, 0, 0, 0, 0, 0};
    __builtin_amdgcn_tensor_load_to_lds(g0, g1, z4, z4, z8, 0);
}

// ---------------- degree / inverse degree ----------------------------------
__global__ void degree_kernel(const int* __restrict__ dst, float* __restrict__ deg, int E) {
    int e = blockIdx.x * blockDim.x + threadIdx.x;
    if (e < E) unsafeAtomicAdd(&deg[dst[e]], 1.0f);
}

__global__ void inv_deg_kernel(float* __restrict__ deg, int n) {
    int i = blockIdx.x * blockDim.x + threadIdx.x;
    if (i < n) deg[i] = 1.0f / fmaxf(deg[i], 1.0f);
}

// ---------------- edge scatter: AGG[dst] += X[src] --------------------------
// float4 row reads (global_load_b128) + 4 hardware float atomics per thread.
// K/4 threads per edge, 256 threads per block.
template <int K>
__global__ void scatter_add_kernel(const float* __restrict__ X,
                                   const int* __restrict__ src,
                                   const int* __restrict__ dst,
                                   float* __restrict__ AGG) {
    constexpr int TPE = K / 4;            // threads per edge
    constexpr int EPB = 256 / TPE;        // edges per block
    const int el = threadIdx.x / TPE;
    const int f4 = (threadIdx.x % TPE) * 4;
    const long e = (long)blockIdx.x * EPB + el;
    const int s = src[e];
    const int d = dst[e];
    const float4 v = *(const float4*)(X + (long)s * K + f4);
    float* p = AGG + (long)d * K + f4;
    unsafeAtomicAdd(p + 0, v.x);
    unsafeAtomicAdd(p + 1, v.y);
    unsafeAtomicAdd(p + 2, v.z);
    unsafeAtomicAdd(p + 3, v.w);
}

// ---------------- fused SAGE GEMM on fp32 WMMA ------------------------------
// C[M x NOUT] = (AGG * inv_deg[row]) @ Wl + X @ Wr + bias
// One block = one 16-row strip; NOUT/16 waves, one 16x16 column tile each.
// Wave 0 DMA-stages both 16xK A panels into LDS with the TDM; all waves then
// run the WMMA K-loop from LDS (A) and L2-hot weights (B).
// Fragments follow the ISA f32 WMMA layouts:
//   A 16x4: lanes 0-15 hold M=0..15 {K=0,1}; lanes 16-31 {K=2,3}
//   B 4x16: lanes 0-15 hold N=0..15 {K=0,1}; lanes 16-31 {K=2,3}
//   C/D   : 8 VGPRs; lanes<16 -> M=vgpr, lanes>=16 -> M=vgpr+8; N=lane&15
template <int K, int NOUT>
__global__ __launch_bounds__((NOUT / 16) * 32)
void sage_gemm(const float* __restrict__ AGG, const float* __restrict__ inv_deg,
               const float* __restrict__ X,
               const float* __restrict__ Wl, const float* __restrict__ Wr,
               const float* __restrict__ bias, float* __restrict__ C) {
    __shared__ float sA1[16 * K];   // agg panel
    __shared__ float sA2[16 * K];   // x   panel

    const int lane = threadIdx.x & 31;
    const int wv   = threadIdx.x >> 5;
    const int m    = lane & 15;
    const int kh   = (lane >> 4) << 1;          // 0 or 2 (half-wave K pair)
    const long row0 = (long)blockIdx.x * 16;

    if (threadIdx.x < 32) {                     // wave 0 drives the TDM
        tdm_load_2d(sA1, AGG + row0 * K, K, 16, K);
        tdm_load_2d(sA2, X   + row0 * K, K, 16, K);
        __builtin_amdgcn_s_wait_tensorcnt(0);
    }
    __syncthreads();

    const float idg = inv_deg[row0 + m];
    const int col = wv * 16 + m;
    const float* __restrict__ wl = Wl + col;
    const float* __restrict__ wr = Wr + col;

    v8f acc = {};
#pragma unroll 4
    for (int k = 0; k < K; k += 4) {
        const float2 A1 = *(const float2*)&sA1[m * K + k + kh];  // ds_load_b64
        const float2 A2 = *(const float2*)&sA2[m * K + k + kh];
        v2f a1; a1.x = A1.x * idg; a1.y = A1.y * idg;
        v2f a2; a2.x = A2.x;       a2.y = A2.y;
        v2f bl; bl.x = wl[(k + kh) * NOUT]; bl.y = wl[(k + kh + 1) * NOUT];
        v2f br; br.x = wr[(k + kh) * NOUT]; br.y = wr[(k + kh + 1) * NOUT];
        acc = __builtin_amdgcn_wmma_f32_16x16x4_f32(false, a1, false, bl,
                                                    (short)0, acc, false, false);
        acc = __builtin_amdgcn_wmma_f32_16x16x4_f32(false, a2, false, br,
                                                    (short)0, acc, false, false);
    }

    const float b = bias[col];
    const int  hi = (lane >> 4) << 3;           // 0 or 8
#pragma unroll
    for (int i = 0; i < 8; ++i)
        C[(row0 + hi + i) * NOUT + col] = acc[i] + b;
}

// ---------------- BatchNorm (training batch stats) --------------------------
__global__ void col_stats_kernel(const float* __restrict__ H, int M,
                                 float* __restrict__ sum, float* __restrict__ sumsq) {
    const int D = blockDim.x;
    const int c = threadIdx.x;
    float s = 0.0f, q = 0.0f;
    for (int r = blockIdx.x; r < M; r += gridDim.x) {
        float v = H[(long)r * D + c];
        s += v;
        q  = fmaf(v, v, q);
    }
    unsafeAtomicAdd(&sum[c], s);
    unsafeAtomicAdd(&sumsq[c], q);
}

__global__ void bn_finalize_kernel(const float* __restrict__ sum,
                                   const float* __restrict__ sumsq,
                                   const float* __restrict__ gamma,
                                   const float* __restrict__ beta,
                                   float* __restrict__ scale,
                                   float* __restrict__ shift, float invM) {
    int c = threadIdx.x;
    float m  = sum[c] * invM;
    float v  = sumsq[c] * invM - m * m;          // biased variance
    float sc = gamma[c] * rsqrtf(v + BN_EPS);
    scale[c] = sc;
    shift[c] = fmaf(-m, sc, beta[c]);
}

// float4-vectorized in-place normalize + ReLU
__global__ void bn_relu_kernel(float* __restrict__ H,
                               const float* __restrict__ scale,
                               const float* __restrict__ shift,
                               int Dmask, long total4) {
    long idx    = (long)blockIdx.x * blockDim.x + threadIdx.x;
    long stride = (long)gridDim.x * blockDim.x;
    for (; idx < total4; idx += stride) {
        long e = idx << 2;
        int  c = (int)(e & Dmask);
        float4 v = *(float4*)(H + e);
        v.x = fmaxf(fmaf(v.x, scale[c + 0], shift[c + 0]), 0.0f);
        v.y = fmaxf(fmaf(v.y, scale[c + 1], shift[c + 1]), 0.0f);
        v.z = fmaxf(fmaf(v.z, scale[c + 2], shift[c + 2]), 0.0f);
        v.w = fmaxf(fmaf(v.w, scale[c + 3], shift[c + 3]), 0.0f);
        *(float4*)(H + e) = v;
    }
}

// ---------------- head: out = h3 @ wh + bh ----------------------------------
__global__ void head_kernel(const float* __restrict__ H, const float* __restrict__ wh,
                            const float* __restrict__ bh, float* __restrict__ out, int M) {
    const int lane = threadIdx.x & 31;
    const int wave = threadIdx.x >> 5;
    const int row  = blockIdx.x * (blockDim.x >> 5) + wave;
    if (row >= M) return;
    const float* hr = H + (long)row * 64;
    float v = fmaf(hr[lane], wh[lane], hr[lane + 32] * wh[lane + 32]);
#pragma unroll
    for (int off = 16; off > 0; off >>= 1) v += __shfl_down(v, off, 32);
    if (lane == 0) out[row] = v + bh[0];
}

// ---------------------------------------------------------------------------
extern "C" void kernel_launch(void* const* d_in, const int* in_sizes, int n_in,
                              void* d_out, int out_size, void* d_ws, size_t ws_size,
                              hipStream_t stream) {
    const float* x   = (const float*)d_in[0];
    const int*   ei  = (const int*)d_in[1];     // int32 edge_index [2,E]
    const float* w1l = (const float*)d_in[2];
    const float* b1l = (const float*)d_in[3];
    const float* w1r = (const float*)d_in[4];
    const float* g1  = (const float*)d_in[5];
    const float* be1 = (const float*)d_in[6];
    const float* w2l = (const float*)d_in[7];
    const float* b2l = (const float*)d_in[8];
    const float* w2r = (const float*)d_in[9];
    const float* g2  = (const float*)d_in[10];
    const float* be2 = (const float*)d_in[11];
    const float* w3l = (const float*)d_in[12];
    const float* b3l = (const float*)d_in[13];
    const float* w3r = (const float*)d_in[14];
    const float* g3  = (const float*)d_in[15];
    const float* be3 = (const float*)d_in[16];
    const float* wh  = (const float*)d_in[17];
    const float* bh  = (const float*)d_in[18];
    float* out = (float*)d_out;

    const int E = in_sizes[1] / 2;
    const int N = NN;
    const int* src = ei;
    const int* dst = ei + E;

    // workspace layout (floats): ~128.2 MB total
    float* ws    = (float*)d_ws;
    float* deg   = ws;                  // N (becomes inv_deg in place)
    float* sum   = ws + 50048;          // 256
    float* sumsq = sum + 256;           // 256
    float* scale = sum + 512;           // 256
    float* shift = sum + 768;           // 256
    float* agg   = ws + 51072;          // N*256 (reused each layer)
    float* h1    = ws + 12851072;       // N*256 (h3 aliases this region)
    float* h2    = ws + 25651072;       // N*128
    float* h3    = h1;                  // N*64, h1 dead by then

    const int MT = N / 16;              // 3125 row strips, exact

    // ---- degrees -> inverse degrees
    (void)hipMemsetAsync(deg, 0, (size_t)N * sizeof(float), stream);
    degree_kernel<<<(E + 255) / 256, 256, 0, stream>>>(dst, deg, E);
    inv_deg_kernel<<<(N + 255) / 256, 256, 0, stream>>>(deg, N);

    // ---- layer 1: 256 -> 256
    (void)hipMemsetAsync(agg, 0, (size_t)N * 256 * sizeof(float), stream);
    scatter_add_kernel<256><<<E / 4, 256, 0, stream>>>(x, src, dst, agg);
    sage_gemm<256, 256><<<MT, 512, 0, stream>>>(agg, deg, x, w1l, w1r, b1l, h1);
    (void)hipMemsetAsync(sum, 0, 512 * sizeof(float), stream);
    col_stats_kernel<<<512, 256, 0, stream>>>(h1, N, sum, sumsq);
    bn_finalize_kernel<<<1, 256, 0, stream>>>(sum, sumsq, g1, be1, scale, shift, 1.0f / N);
    bn_relu_kernel<<<2048, 256, 0, stream>>>(h1, scale, shift, 255, (long)N * 64);

    // ---- layer 2: 256 -> 128
    (void)hipMemsetAsync(agg, 0, (size_t)N * 256 * sizeof(float), stream);
    scatter_add_kernel<256><<<E / 4, 256, 0, stream>>>(h1, src, dst, agg);
    sage_gemm<256, 128><<<MT, 256, 0, stream>>>(agg, deg, h1, w2l, w2r, b2l, h2);
    (void)hipMemsetAsync(sum, 0, 512 * sizeof(float), stream);
    col_stats_kernel<<<512, 128, 0, stream>>>(h2, N, sum, sumsq);
    bn_finalize_kernel<<<1, 128, 0, stream>>>(sum, sumsq, g2, be2, scale, shift, 1.0f / N);
    bn_relu_kernel<<<2048, 256, 0, stream>>>(h2, scale, shift, 127, (long)N * 32);

    // ---- layer 3: 128 -> 64  (h3 reuses h1's region; h1 is dead now)
    (void)hipMemsetAsync(agg, 0, (size_t)N * 128 * sizeof(float), stream);
    scatter_add_kernel<128><<<E / 8, 256, 0, stream>>>(h2, src, dst, agg);
    sage_gemm<128, 64><<<MT, 128, 0, stream>>>(agg, deg, h2, w3l, w3r, b3l, h3);
    (void)hipMemsetAsync(sum, 0, 512 * sizeof(float), stream);
    col_stats_kernel<<<512, 64, 0, stream>>>(h3, N, sum, sumsq);
    bn_finalize_kernel<<<1, 64, 0, stream>>>(sum, sumsq, g3, be3, scale, shift, 1.0f / N);
    bn_relu_kernel<<<2048, 256, 0, stream>>>(h3, scale, shift, 63, (long)N * 16);

    // ---- head
    head_kernel<<<(N + 7) / 8, 256, 0, stream>>>(h3, wh, bh, out, N);
}